// EdgeConv_55155970015771
// MI455X (gfx1250) — compile-verified
//
#include <hip/hip_runtime.h>
#include <math.h>

// EdgeConv: B=16, N=8192, K=20, D=64. out = GELU(LN(max_k((nx-c)W1 + c W2)))
// Restructured: y = x*A^T, z = x*(B-A)^T  (one fp32 WMMA GEMM), then
// out = LN_GELU(max_k y[ind_k] + z). Gathered y (33.5MB) is L2-resident.
//
// Workspace layout: [ WcT : 128*64 f32 = 32KB ][ t : 131072*128 f32 = 64MiB ]

#define BB 16
#define NN 8192
#define KK 20
#define DD 64
#define MM (BB * NN)          // 131072 rows
#define TCOLS 128             // y (64) | z (64)

typedef __attribute__((ext_vector_type(2))) float v2f;
typedef __attribute__((ext_vector_type(8))) float v8f;

// ---------------------------------------------------------------------------
// Kernel 0: build combined transposed weight WcT[j][d], j in [0,128), d in [0,64)
//   j <  64 : WcT[j][d] = W[j][d]              (A row, applied to neighbor)
//   j >= 64 : WcT[j][d] = W[j-64][64+d] - W[j-64][d]   ((B-A) row, applied to center)
__global__ void __launch_bounds__(256) edgeconv_prep_wct(
    const float* __restrict__ W, float* __restrict__ WcT) {
  int tid = blockIdx.x * 256 + threadIdx.x;   // 0..8191
  int j = tid >> 6;
  int d = tid & 63;
  float v;
  if (j < 64) {
    v = W[j * 128 + d];
  } else {
    int o = j - 64;
    v = W[o * 128 + 64 + d] - W[o * 128 + d];
  }
  WcT[j * 64 + d] = v;
}

// ---------------------------------------------------------------------------
// Kernel 1: t[m][0:128] = x[m][0:64] * WcT^T via V_WMMA_F32_16X16X4_F32.
// One wave per 16-row M-tile; all 8 N-tiles (128 cols) per wave, K=64 in 16 steps.
// A 16x4 f32 frag layout: lanes 0-15 -> M=lane, K={4k,4k+1}; lanes 16-31 -> K={4k+2,4k+3}.
// B 4x16 frag layout (mirror): lane%16 -> N, lane<16 -> K={4k,4k+1}, else K={4k+2,4k+3}.
// C/D 16x16 f32: VGPR r -> M = r + (lane>=16 ? 8 : 0), N = lane%16.
__global__ void __launch_bounds__(256) edgeconv_gemm_wmma(
    const float* __restrict__ x,      // [MM][64]
    const float* __restrict__ WcT,    // [128][64]
    float* __restrict__ t) {          // [MM][128]
  __shared__ float sW[TCOLS * DD];    // 32 KB of the 320 KB LDS
  // Cooperative stage of WcT into LDS (128B / thread as float4s).
  for (int i = threadIdx.x; i < (TCOLS * DD) / 4; i += 256) {
    ((float4*)sW)[i] = ((const float4*)WcT)[i];
  }
  __syncthreads();

  const int wave = threadIdx.x >> 5;
  const int lane = threadIdx.x & 31;
  const int half = lane >> 4;             // 0: K lo pair, 1: K hi pair
  const int l    = lane & 15;
  const int mTile = blockIdx.x * 8 + wave;   // 0..8191, exact cover: no divergence
  const int m0 = mTile * 16;

  const float* xrow = x + (size_t)(m0 + l) * DD;   // A-frag row (M = l)

  v8f acc[8] = {};                         // 8 N-tiles of 16x16 f32 accumulators
  #pragma unroll
  for (int k = 0; k < 16; ++k) {           // K = 64 in steps of 4
    v2f a = *(const v2f*)(xrow + k * 4 + half * 2);
    #pragma unroll
    for (int j = 0; j < 8; ++j) {
      v2f bf = *(const v2f*)(&sW[(j * 16 + l) * DD + k * 4 + half * 2]);
      acc[j] = __builtin_amdgcn_wmma_f32_16x16x4_f32(
          /*neg_a=*/false, a, /*neg_b=*/false, bf,
          /*c_mod=*/(short)0, acc[j], /*reuse_a=*/false, /*reuse_b=*/false);
    }
  }

  // Store D tiles.
  #pragma unroll
  for (int j = 0; j < 8; ++j) {
    float* base = t + (size_t)(m0 + half * 8) * TCOLS + j * 16 + l;
    #pragma unroll
    for (int r = 0; r < 8; ++r) {
      base[(size_t)r * TCOLS] = acc[j][r];
    }
  }
}

// ---------------------------------------------------------------------------
// Kernel 2: per row m: hmax[o] = max_k y[b, ind[k]][o]; h = hmax + z[m];
// LayerNorm over 64, exact GELU. One wave per row, 2 columns per lane.
__global__ void __launch_bounds__(256) edgeconv_gather_ln_gelu(
    const float* __restrict__ t,        // [MM][128] : y | z
    const int*   __restrict__ ind,      // [MM][20]
    const float* __restrict__ gamma,    // [64]
    const float* __restrict__ beta,     // [64]
    float*       __restrict__ out) {    // [MM][64]
  const int wave = threadIdx.x >> 5;
  const int lane = threadIdx.x & 31;
  const int m = blockIdx.x * 8 + wave;     // 0..131071, exact cover
  const int b = m >> 13;                   // m / N  (N = 8192)
  const int o = lane * 2;

  // Lanes 0..19 fetch the 20 indices once; broadcast via readlane shuffles.
  const int* idxp = ind + (size_t)m * KK;
  int myidx = (lane < KK) ? idxp[lane] : 0;

  float hx = -INFINITY, hy = -INFINITY;
  #pragma unroll
  for (int k = 0; k < KK; ++k) {
    int idx = __shfl(myidx, k, 32);                       // uniform broadcast
    const float* yrow = t + ((size_t)(b * NN + idx)) * TCOLS;
    float2 yv = *(const float2*)(yrow + o);               // 256B/row, L2-resident
    hx = fmaxf(hx, yv.x);
    hy = fmaxf(hy, yv.y);
  }
  float2 zv = *(const float2*)(t + (size_t)m * TCOLS + DD + o);
  hx += zv.x;
  hy += zv.y;

  // LayerNorm over 64 values (2 per lane, wave32 butterfly reductions).
  float s = hx + hy;
  #pragma unroll
  for (int off = 16; off > 0; off >>= 1) s += __shfl_xor(s, off, 32);
  const float mu = s * (1.0f / 64.0f);
  const float dx = hx - mu, dy = hy - mu;
  float v = dx * dx + dy * dy;
  #pragma unroll
  for (int off = 16; off > 0; off >>= 1) v += __shfl_xor(v, off, 32);
  const float rstd = rsqrtf(v * (1.0f / 64.0f) + 1e-5f);

  float n0 = dx * rstd * gamma[o]     + beta[o];
  float n1 = dy * rstd * gamma[o + 1] + beta[o + 1];

  // Exact GELU: 0.5*x*(1+erf(x/sqrt(2)))
  const float is2 = 0.70710678118654752440f;
  float o0 = 0.5f * n0 * (1.0f + erff(n0 * is2));
  float o1 = 0.5f * n1 * (1.0f + erff(n1 * is2));

  *(float2*)(out + (size_t)m * DD + o) = make_float2(o0, o1);
}

// ---------------------------------------------------------------------------
extern "C" void kernel_launch(void* const* d_in, const int* in_sizes, int n_in,
                              void* d_out, int out_size, void* d_ws, size_t ws_size,
                              hipStream_t stream) {
  const float* x     = (const float*)d_in[0];
  const int*   ind   = (const int*)d_in[1];
  const float* W     = (const float*)d_in[2];
  const float* gamma = (const float*)d_in[3];
  const float* beta  = (const float*)d_in[4];
  float* out = (float*)d_out;

  float* WcT = (float*)d_ws;              // 32 KB
  float* t   = WcT + TCOLS * DD;          // 64 MiB: [MM][128]
  (void)ws_size; (void)in_sizes; (void)n_in; (void)out_size;

  // 8192 WcT elements / 256 threads
  edgeconv_prep_wct<<<32, 256, 0, stream>>>(W, WcT);
  // 8192 M-tiles / 8 waves per block
  edgeconv_gemm_wmma<<<1024, 256, 0, stream>>>(x, WcT, t);
  // 131072 rows / 8 waves per block
  edgeconv_gather_ln_gelu<<<16384, 256, 0, stream>>>(t, ind, gamma, beta, out);
}